// MultiHeadAttention2D_71734543778255
// MI455X (gfx1250) — compile-verified
//
#include <hip/hip_runtime.h>
#include <hip/hip_bf16.h>
#include <math.h>

// Multi-head attention (B=4, C=256, H=W=48, heads=8, d_k=32) with residual.
// Flash-attention streaming softmax; all GEMMs on V_WMMA_F32_16X16X4_F32.

typedef __attribute__((ext_vector_type(2))) float v2f;
typedef __attribute__((ext_vector_type(8))) float v8f;

#define B_   4
#define C_   256
#define HDS  8
#define DK   32
#define N_   2304
#define NT   (N_ / 16)          // 144 tiles along N
#define QSCALE 0.17677669529663687f  // 1/sqrt(32)

static __device__ __forceinline__ v8f wmma_f32(v2f a, v2f b, v8f c) {
  // D = A(16x4) * B(4x16) + C(16x16), full fp32
  return __builtin_amdgcn_wmma_f32_16x16x4_f32(
      false, a, false, b, (short)0, c, false, false);
}

// ---------------------------------------------------------------------------
// Projections: per batch  W(256x256) @ X(256x2304).
// One wave -> one 16x16 output tile, K swept 4 at a time (64 WMMAs).
// p==0 -> Qt[b,h,n,d] (pre-scaled by 1/sqrt(dk))
// p==1 -> K  [b,c,n]  (natural layout, B-matrix friendly for Q^T K)
// p==2 -> Vt[b,h,n,d] (B-matrix friendly for P @ V^T)
// ---------------------------------------------------------------------------
__global__ __launch_bounds__(128) void mha_proj_kernel(
    const float* __restrict__ x,
    const float* __restrict__ wq, const float* __restrict__ wk,
    const float* __restrict__ wv,
    float* __restrict__ qt, float* __restrict__ kb, float* __restrict__ vt) {
  const int lane = threadIdx.x & 31;
  const int wave = threadIdx.x >> 5;
  const int t     = blockIdx.x * 4 + wave;   // 0..2303 tiles
  const int oTile = t & 15;                  // 16 tiles over out-channels
  const int nTile = t >> 4;                  // 144 tiles over N
  const int p = blockIdx.y % 3;              // 0=q 1=k 2=v
  const int b = blockIdx.y / 3;
  const float* w = (p == 0) ? wq : (p == 1) ? wk : wv;

  const int row  = lane & 15;                // M (lanes 0-15) / shared col idx
  const int half = lane >> 4;
  const int koff = half * 2;                 // A/B: VGPR0 holds K{0,2}, VGPR1 K{1,3}
  const int oBase = oTile * 16;
  const int nBase = nTile * 16;

  const float* wA = w + (size_t)(oBase + row) * C_ + koff;
  const float* xB = x + ((size_t)b * C_ + koff) * N_ + nBase + row;

  v8f acc = {};
#pragma unroll 8
  for (int k = 0; k < C_; k += 4) {
    v2f a, bm;
    a.x  = wA[k];
    a.y  = wA[k + 1];
    bm.x = xB[(size_t)k * N_];
    bm.y = xB[(size_t)(k + 1) * N_];
    acc = wmma_f32(a, bm, acc);
  }

  if (p == 1) {
    // natural [b, c, n]; C-layout: VGPR v -> row oBase+v+8*half, col nBase+row
    float* outp = kb + ((size_t)b * C_ + oBase + half * 8) * N_ + nBase + row;
#pragma unroll
    for (int v = 0; v < 8; ++v) outp[(size_t)v * N_] = acc[v];
  } else {
    float* dst = (p == 0) ? qt : vt;
    const float scale = (p == 0) ? QSCALE : 1.0f;
#pragma unroll
    for (int v = 0; v < 8; ++v) {
      const int o = oBase + v + half * 8;
      const int h = o >> 5, d = o & 31;
      dst[(((size_t)b * HDS + h) * N_ + nBase + row) * DK + d] = acc[v] * scale;
    }
  }
}

// ---------------------------------------------------------------------------
// Flash attention + residual. One wave per (b, h, 16-query-row tile).
// ---------------------------------------------------------------------------
__global__ __launch_bounds__(128) void mha_attn_kernel(
    const float* __restrict__ x,
    const float* __restrict__ qt, const float* __restrict__ kb,
    const float* __restrict__ vt, float* __restrict__ out) {
  __shared__ float ldsP[4][16 * 17];   // padded 16x16 P tile per wave
  const int lane = threadIdx.x & 31;
  const int wave = threadIdx.x >> 5;
  float* P = ldsP[wave];

  const int task  = blockIdx.x * 4 + wave;   // 0..4607
  const int qTile = task % NT;
  const int bh    = task / NT;               // 0..31
  const int h = bh & 7;
  const int b = bh >> 3;

  const int row  = lane & 15;
  const int half = lane >> 4;
  const int koff = half * 2;
  const int nBase = qTile * 16;

  // Preload the whole Q fragment: A(M=query row, K=d), 16x32 -> 8 steps of K=4
  v2f aq[8];
  {
    const float* qp =
        qt + (((size_t)b * HDS + h) * N_ + nBase + row) * DK + koff;
#pragma unroll
    for (int kk = 0; kk < 8; ++kk) {
      aq[kk].x = qp[kk * 4];
      aq[kk].y = qp[kk * 4 + 1];
    }
  }

  const float* kbase = kb + ((size_t)b * C_ + h * DK + koff) * N_ + row;
  const float* vhead = vt + ((size_t)b * HDS + h) * N_ * DK;

  v8f o0 = {}, o1 = {};              // O tile: rows=query n, cols=d (0-15 / 16-31)
  float m[8], l[8];
#pragma unroll
  for (int v = 0; v < 8; ++v) { m[v] = -1e30f; l[v] = 0.0f; }

  for (int mt = 0; mt < NT; ++mt) {
    const int mBase = mt * 16;

    // S = Q^T K  (16 query rows x 16 key cols), K-dim = d_k = 32
    v8f S = {};
#pragma unroll
    for (int kk = 0; kk < 8; ++kk) {
      const float* kp = kbase + (size_t)(kk * 4) * N_ + mBase;
      v2f bk;
      bk.x = kp[0];
      bk.y = kp[N_];
      S = wmma_f32(aq[kk], bk, S);
    }

    // Online softmax. C-layout: VGPR v + lane-half <-> row v+8*half, 16 cols
    // of a row live on one 16-lane half -> xor-reduce with masks 1,2,4,8.
#pragma unroll
    for (int v = 0; v < 8; ++v) {
      float s  = S[v];
      float rm = s;
      rm = fmaxf(rm, __shfl_xor(rm, 1, 16));
      rm = fmaxf(rm, __shfl_xor(rm, 2, 16));
      rm = fmaxf(rm, __shfl_xor(rm, 4, 16));
      rm = fmaxf(rm, __shfl_xor(rm, 8, 16));
      const float mn = fmaxf(m[v], rm);
      const float a  = __expf(m[v] - mn);
      const float pv = __expf(s - mn);
      float rs = pv;
      rs += __shfl_xor(rs, 1, 16);
      rs += __shfl_xor(rs, 2, 16);
      rs += __shfl_xor(rs, 4, 16);
      rs += __shfl_xor(rs, 8, 16);
      l[v] = l[v] * a + rs;
      m[v] = mn;
      o0[v] *= a;
      o1[v] *= a;
      P[(v + half * 8) * 17 + row] = pv;   // transpose staging (C -> A layout)
    }
    asm volatile("s_wait_dscnt 0" ::: "memory");  // P tile visible to own wave

    // O += P(16xn,16 m) * V^T(16 m, 32 d); A from LDS, K-dim = 16 -> 4 steps
#pragma unroll
    for (int kk = 0; kk < 4; ++kk) {
      v2f ap;
      ap.x = P[row * 17 + kk * 4 + koff];
      ap.y = P[row * 17 + kk * 4 + koff + 1];
      const float* vp = vhead + (size_t)(mBase + kk * 4 + koff) * DK + row;
      v2f bv0, bv1;
      bv0.x = vp[0];
      bv0.y = vp[DK];
      bv1.x = vp[16];
      bv1.y = vp[DK + 16];
      o0 = wmma_f32(ap, bv0, o0);
      o1 = wmma_f32(ap, bv1, o1);
    }
  }

  // Epilogue: divide by softmax denominator, fuse residual, write out.
  const size_t obase =
      ((size_t)b * C_ + h * DK + row) * N_ + nBase + half * 8;
#pragma unroll
  for (int v = 0; v < 8; ++v) {
    const float inv = 1.0f / l[v];
    const size_t i0 = obase + v;              // d = row (cols 0-15)
    out[i0] = x[i0] + o0[v] * inv;
    const size_t i1 = i0 + (size_t)16 * N_;   // d = row + 16
    out[i1] = x[i1] + o1[v] * inv;
  }
}

// ---------------------------------------------------------------------------
extern "C" void kernel_launch(void* const* d_in, const int* in_sizes, int n_in,
                              void* d_out, int out_size, void* d_ws,
                              size_t ws_size, hipStream_t stream) {
  const float* x  = (const float*)d_in[0];
  const float* wq = (const float*)d_in[1];
  const float* wk = (const float*)d_in[2];
  const float* wv = (const float*)d_in[3];
  float* out = (float*)d_out;

  // Workspace: Qt | K | Vt, each B*C*N floats (3 * 9.44 MB = 28.3 MB)
  float* qt = (float*)d_ws;
  float* kb = qt + (size_t)B_ * C_ * N_;
  float* vt = kb + (size_t)B_ * C_ * N_;

  dim3 blk(128);
  // 2304 tiles per (b,proj), 4 waves/block -> 576 blocks; y = b*3+p
  mha_proj_kernel<<<dim3(576, 12), blk, 0, stream>>>(x, wq, wk, wv, qt, kb, vt);
  // 4*8*144 = 4608 wave-tasks, 4 waves/block -> 1152 blocks
  mha_attn_kernel<<<dim3(1152), blk, 0, stream>>>(x, qt, kb, vt, out);
}